// ChamferLoss_22797686407325
// MI455X (gfx1250) — compile-verified
//
#include <hip/hip_runtime.h>

typedef __attribute__((ext_vector_type(2))) float v2f;
typedef __attribute__((ext_vector_type(8))) float v8f;

#define KPTS 2048     // points per set per batch
#define DTOT 4096     // floats per batch row (2*KPTS)
#define NBATCH 64

// One block = 8 waves, one batch, 8 row-blocks of A (16 rows each).
// Each wave scans all 128 column tiles of B via V_WMMA_F32_16X16X4_F32,
// where A-rows carry (|p|^2, 1, px, py) and B-cols carry (1, |t|^2, -2tx, -2ty),
// so D = A x B is the 16x16 tile of squared distances.
//
// B-fragments are precomputed once per block into LDS in the exact per-lane
// WMMA layout, so the inner loop is just: ds_load_b64 -> v_wmma -> v_min3.
__global__ __launch_bounds__(256) void chamfer_dir_kernel(
    const float* __restrict__ Apts, const float* __restrict__ Bpts,
    float* __restrict__ partials, int dirOffset)
{
    // [0][i]: K={0,1} lanes (lanes 0-15): (1, |t_i|^2)
    // [1][i]: K={2,3} lanes (lanes 16-31): (-2*t_ix, -2*t_iy)
    __shared__ v2f sBfrag[2][KPTS];   // 32 KB
    __shared__ float wsum[8];

    const int b   = blockIdx.y;
    const int tid = threadIdx.x;
    const float* Ab = Apts + (size_t)b * DTOT;
    const float* Bb = Bpts + (size_t)b * DTOT;

    // Build all 2048 B-point fragments for this batch once, coalesced.
    for (int i = tid; i < KPTS; i += 256) {
        const float tx = Bb[i];
        const float ty = Bb[KPTS + i];
        v2f f0; f0.x = 1.0f;         f0.y = tx * tx + ty * ty;
        v2f f1; f1.x = -2.0f * tx;   f1.y = -2.0f * ty;
        sBfrag[0][i] = f0;
        sBfrag[1][i] = f1;
    }
    __syncthreads();

    const int wave    = tid >> 5;
    const int lane    = tid & 31;
    const int halfSel = lane >> 4;   // 0: lanes 0-15 carry K={0,1}; 1: lanes 16-31 carry K={2,3}
    const int ml      = lane & 15;

    const int rowblock = blockIdx.x * 8 + wave;   // 0..127
    const int m  = rowblock * 16 + ml;            // A point index for this lane's M
    const float px = Ab[m];
    const float py = Ab[KPTS + m];

    // A fragment (16x4 f32, M x K): lane layout per CDNA5 ISA 7.12.2
    v2f afrag;
    afrag.x = halfSel ? px : (px * px + py * py);  // K0 or K2
    afrag.y = halfSel ? py : 1.0f;                 // K1 or K3

    float rmin[8];
#pragma unroll
    for (int r = 0; r < 8; ++r) rmin[r] = 3.0e38f;

    const v8f czero = {0.f, 0.f, 0.f, 0.f, 0.f, 0.f, 0.f, 0.f};
    const v2f* __restrict__ bRow = &sBfrag[halfSel][ml];

#pragma unroll 4
    for (int j = 0; j < KPTS / 16; ++j) {
        const v2f bfrag = bRow[j * 16];   // one ds_load_b64 per tile

        // 8 args: (neg_a, A, neg_b, B, c_mod, C, reuse_a, reuse_b)
        v8f d = __builtin_amdgcn_wmma_f32_16x16x4_f32(
            false, afrag, false, bfrag, (short)0, czero, false, false);

#pragma unroll
        for (int r = 0; r < 8; ++r) rmin[r] = fminf(rmin[r], d[r]);
    }

    // Row-min across the 16 lanes within each half (same M per {r, half}).
#pragma unroll
    for (int r = 0; r < 8; ++r) {
        float v = rmin[r];
        v = fminf(v, __shfl_xor(v, 1, 32));
        v = fminf(v, __shfl_xor(v, 2, 32));
        v = fminf(v, __shfl_xor(v, 4, 32));
        v = fminf(v, __shfl_xor(v, 8, 32));
        rmin[r] = v;
    }

    // sqrt(max(.,0)) once per row, sum the 8 rows this lane-half owns.
    float s = 0.0f;
#pragma unroll
    for (int r = 0; r < 8; ++r) s += sqrtf(fmaxf(rmin[r], 0.0f));
    // half0 holds rows 0-7 sum, half1 holds rows 8-15 sum -> combine.
    s += __shfl_xor(s, 16, 32);

    if (lane == 0) wsum[wave] = s;
    __syncthreads();
    if (tid == 0) {
        float t = 0.0f;
        for (int w = 0; w < 8; ++w) t += wsum[w];
        partials[dirOffset + blockIdx.y * 16 + blockIdx.x] = t;
    }
}

// Deterministic final reduction of 2048 per-block partials -> scalar mean.
__global__ __launch_bounds__(256) void chamfer_reduce_kernel(
    const float* __restrict__ partials, float* __restrict__ out)
{
    __shared__ float acc[256];
    const int tid = threadIdx.x;
    float s = 0.0f;
    for (int i = tid; i < 2048; i += 256) s += partials[i];  // fixed order
    acc[tid] = s;
    __syncthreads();
    if (tid == 0) {
        float t = 0.0f;
        for (int i = 0; i < 256; ++i) t += acc[i];           // fixed order
        out[0] = t * (1.0f / ((float)NBATCH * (float)KPTS));
    }
}

extern "C" void kernel_launch(void* const* d_in, const int* in_sizes, int n_in,
                              void* d_out, int out_size, void* d_ws, size_t ws_size,
                              hipStream_t stream) {
    const float* pred = (const float*)d_in[0];
    const float* targ = (const float*)d_in[1];
    float* out = (float*)d_out;
    float* ws  = (float*)d_ws;   // needs 2048 floats (8 KB)

    dim3 block(256);
    dim3 grid(16, NBATCH);       // 16 row-block groups x 64 batches

    // Pass 1: for each predicted point, min over targets (min2 term).
    chamfer_dir_kernel<<<grid, block, 0, stream>>>(pred, targ, ws, 0);
    // Pass 2: for each target point, min over predicted (min1 term).
    chamfer_dir_kernel<<<grid, block, 0, stream>>>(targ, pred, ws, 1024);
    // Combine: mean over batch of (min1+min2) == sum(all mins)/(64*2048).
    chamfer_reduce_kernel<<<dim3(1), block, 0, stream>>>(ws, out);
}